// RotationInvariantEncoding_90305982365990
// MI455X (gfx1250) — compile-verified
//
#include <hip/hip_runtime.h>
#include <stdint.h>

// ---------------------------------------------------------------------------
// ORN Rotation-Invariant Encoding for MI455X (gfx1250, wave32).
// x: [N, F, O=8] fp32.  d = argmax_o x[n,f,o];  out[n,f,o] = x[n,f,(o+d)%8].
// Memory-bound: 512 MB streamed @ 23.3 TB/s -> ~22 us floor. Strategy:
//   * TDM (tensor_load_to_lds) stages one 8 KB tile per workgroup into LDS
//     (TENSORcnt + s_wait_tensorcnt), issued by wave 0 only.
//   * Each lane owns one 8-float group: 2x ds_load_b128 from LDS,
//     branchless cndmask rotate, 2x non-temporal global_store_b128.
// ---------------------------------------------------------------------------

typedef uint32_t u32x4 __attribute__((ext_vector_type(4)));
typedef int32_t  i32x8 __attribute__((ext_vector_type(8)));
typedef int32_t  i32x4 __attribute__((ext_vector_type(4)));
typedef float    f32x4 __attribute__((ext_vector_type(4)));   // native vector: OK for NT builtins

#ifndef __has_builtin
#define __has_builtin(x) 0
#endif

#if __has_builtin(__builtin_amdgcn_tensor_load_to_lds) && \
    __has_builtin(__builtin_amdgcn_s_wait_tensorcnt)
#define USE_TDM 1
#else
#define USE_TDM 0
#endif

#define ORIENT 8
#define GROUPS_PER_BLOCK 256
#define TILE_FLOATS (GROUPS_PER_BLOCK * ORIENT)   // 2048 floats = 8 KB per WG

__global__ __launch_bounds__(GROUPS_PER_BLOCK)
void orn_rie_kernel(const float* __restrict__ x, float* __restrict__ out,
                    unsigned int numGroups) {
    const unsigned int g = blockIdx.x * GROUPS_PER_BLOCK + threadIdx.x;
    f32x4 a, b;

#if USE_TDM
    __shared__ float tile[TILE_FLOATS];

    const unsigned long long baseElem  = (unsigned long long)blockIdx.x * TILE_FLOATS;
    const unsigned long long totalElem = (unsigned long long)numGroups * ORIENT;
    const unsigned long long remain    = totalElem - baseElem;
    const unsigned int tileElems =
        remain < (unsigned long long)TILE_FLOATS ? (unsigned int)remain : TILE_FLOATS;

    if (threadIdx.x < 32u) {   // exactly wave 0 of this workgroup issues the DMA
        const uint64_t gaddr  = (uint64_t)(uintptr_t)(x + baseElem);
        const uint32_t ldsOff = (uint32_t)(uintptr_t)(&tile[0]);  // low 32b = LDS offset

        // D# group 0 (ISA 8.3): [1:0]=count=1, [63:32]=lds_addr,
        // [120:64]=global_addr, [127:126]=type=2.
        u32x4 d0;
        d0[0] = 1u;
        d0[1] = ldsOff;
        d0[2] = (uint32_t)(gaddr & 0xFFFFFFFFu);
        d0[3] = (uint32_t)((gaddr >> 32) & 0x01FFFFFFu) | 0x80000000u;

        // D# group 1 (ISA 8.4): workgroup_mask=0 (not in cluster),
        // data_size=2 (4 B), tensor_dim0=tile_dim0=tileElems,
        // tensor_dim1=tile_dim1=1, tensor_dim0_stride=tileElems.
        i32x8 d1;
        d1[0] = (int32_t)(2u << 16);                                   // data_size=4B
        d1[1] = (int32_t)((tileElems & 0xFFFFu) << 16);                // tensor_dim0 lo16
        d1[2] = (int32_t)(((tileElems >> 16) & 0xFFFFu) | (1u << 16)); // dim0 hi16 | tensor_dim1=1
        d1[3] = (int32_t)((tileElems & 0xFFFFu) << 16);                // tile_dim0 (<=2048)
        d1[4] = 1;                                                     // tile_dim1=1, tile_dim2=0
        d1[5] = (int32_t)tileElems;                                    // dim0_stride lo32
        d1[6] = 0;
        d1[7] = 0;

        const i32x4 z4 = {0, 0, 0, 0};            // groups 2/3 unused (<=2D tensor)
        const i32x8 z8 = {0, 0, 0, 0, 0, 0, 0, 0};
        __builtin_amdgcn_tensor_load_to_lds(d0, d1, z4, z4, z8, 0);
        __builtin_amdgcn_s_wait_tensorcnt(0);     // tile resident in LDS
    }
    __syncthreads();

    if (g >= numGroups) return;
    const f32x4* lp = reinterpret_cast<const f32x4*>(&tile[threadIdx.x * ORIENT]);
    a = lp[0];
    b = lp[1];
#else
    if (g >= numGroups) return;
    const f32x4* gp = reinterpret_cast<const f32x4*>(x) + (size_t)g * 2;
    a = __builtin_nontemporal_load(gp);           // single-use stream: th=NT
    b = __builtin_nontemporal_load(gp + 1);
#endif

    float v0 = a[0], v1 = a[1], v2 = a[2], v3 = a[3];
    float v4 = b[0], v5 = b[1], v6 = b[2], v7 = b[3];

    // First-occurrence argmax over the 8 orientation channels (strict '>').
    float best = v0; int d = 0;
    if (v1 > best) { best = v1; d = 1; }
    if (v2 > best) { best = v2; d = 2; }
    if (v3 > best) { best = v3; d = 3; }
    if (v4 > best) { best = v4; d = 4; }
    if (v5 > best) { best = v5; d = 5; }
    if (v6 > best) { best = v6; d = 6; }
    if (v7 > best) { best = v7; d = 7; }

    // Branchless rotate-left by d: out[i] = v[(i+d)&7]; 3 cndmask stages.
    const bool r4 = (d & 4) != 0;
    float t0 = r4 ? v4 : v0, t1 = r4 ? v5 : v1, t2 = r4 ? v6 : v2, t3 = r4 ? v7 : v3;
    float t4 = r4 ? v0 : v4, t5 = r4 ? v1 : v5, t6 = r4 ? v2 : v6, t7 = r4 ? v3 : v7;

    const bool r2 = (d & 2) != 0;
    float u0 = r2 ? t2 : t0, u1 = r2 ? t3 : t1, u2 = r2 ? t4 : t2, u3 = r2 ? t5 : t3;
    float u4 = r2 ? t6 : t4, u5 = r2 ? t7 : t5, u6 = r2 ? t0 : t6, u7 = r2 ? t1 : t7;

    const bool r1 = (d & 1) != 0;
    float w0 = r1 ? u1 : u0, w1 = r1 ? u2 : u1, w2 = r1 ? u3 : u2, w3 = r1 ? u4 : u3;
    float w4 = r1 ? u5 : u4, w5 = r1 ? u6 : u5, w6 = r1 ? u7 : u6, w7 = r1 ? u0 : u7;

    f32x4 o0; o0[0] = w0; o0[1] = w1; o0[2] = w2; o0[3] = w3;
    f32x4 o1; o1[0] = w4; o1[1] = w5; o1[2] = w6; o1[3] = w7;
    f32x4* op = reinterpret_cast<f32x4*>(out) + (size_t)g * 2;
    __builtin_nontemporal_store(o0, op);          // streamed once: th=NT, keep L2 clean
    __builtin_nontemporal_store(o1, op + 1);
}

extern "C" void kernel_launch(void* const* d_in, const int* in_sizes, int n_in,
                              void* d_out, int out_size, void* d_ws, size_t ws_size,
                              hipStream_t stream) {
    (void)in_sizes; (void)n_in; (void)d_ws; (void)ws_size;
    const float* x = (const float*)d_in[0];
    // d_in[1] is nOrientation == 8; baked in as compile-time constant (ORIENT)
    // so the rotate network fully unrolls into cndmask chains.
    float* out = (float*)d_out;

    const unsigned int numGroups = (unsigned int)(out_size / ORIENT);   // 8,388,608
    const unsigned int blocks = (numGroups + GROUPS_PER_BLOCK - 1) / GROUPS_PER_BLOCK;

    orn_rie_kernel<<<blocks, GROUPS_PER_BLOCK, 0, stream>>>(x, out, numGroups);
}